// Outflow_70806830841969
// MI455X (gfx1250) — compile-verified
//
#include <hip/hip_runtime.h>
#include <hip/hip_bf16.h>

typedef __attribute__((ext_vector_type(2))) float    v2f;
typedef __attribute__((ext_vector_type(8))) float    v8f;
typedef __attribute__((ext_vector_type(4))) unsigned v4u;
typedef __attribute__((ext_vector_type(8))) int      v8i;
typedef __attribute__((ext_vector_type(4))) int      v4i;

#define F_DIM 128
#define H_DIM 4
#define D_DIM 32
#define LDS_STRIDE 132   // 128 + 4 pad: A-matrix reads hit distinct banks
#define TILE_M 64        // edges per block in edge_kernel

// ---------------------------------------------------------------------------
// One 16x16 fp32 output tile of  A(16x128, LDS) x W(128x128, global).
// W logical element (f, c) lives at  w[(c>>5)*4096 + f*32 + (c&31)]
// because w is stored (H=4, F=128, D=32) and c = h*32 + d.
// WMMA f32 16x16x4 register layouts per CDNA5 ISA 7.12.2.
// ---------------------------------------------------------------------------
__device__ __forceinline__ v8f wmma_gemm_tile(const float* A_lds,
                                              const float* __restrict__ W,
                                              int c0, int lane) {
  const int mrow = lane & 15;
  const int koff = (lane >> 4) << 1;          // 0 for lanes 0-15, 2 for 16-31
  const int col  = c0 + (lane & 15);
  const float* wb = W + ((col >> 5) << 12) + (col & 31);
  v8f acc = {};
#pragma unroll
  for (int kb = 0; kb < 32; ++kb) {
    const int kk = (kb << 2) + koff;
    v2f a, b;
    a.x = A_lds[mrow * LDS_STRIDE + kk];
    a.y = A_lds[mrow * LDS_STRIDE + kk + 1];
    b.x = wb[kk << 5];
    b.y = wb[(kk + 1) << 5];
    acc = __builtin_amdgcn_wmma_f32_16x16x4_f32(false, a, false, b,
                                                (short)0, acc, false, false);
  }
  return acc;
}

// ---------------------------------------------------------------------------
// Kernel 1: init per-(segment,head) state. m_enc uses an order-preserving
// float->u32 encoding so atomicMax(u32) implements float max incl. negatives.
// ---------------------------------------------------------------------------
__global__ void init_seg(unsigned* __restrict__ m_enc, float* __restrict__ denom, int n) {
  int i = blockIdx.x * 256 + threadIdx.x;
  if (i < n) { m_enc[i] = 0u; denom[i] = 0.0f; }
}

// ---------------------------------------------------------------------------
// Kernel 2: q = feat @ w_q, k = feat @ w_k   (N x 128) x (128 x 128) each.
// ---------------------------------------------------------------------------
__global__ void qk_gemm(const float* __restrict__ feat,
                        const float* __restrict__ wq,
                        const float* __restrict__ wk,
                        float* __restrict__ q, float* __restrict__ k, int N) {
  __shared__ float A[16 * LDS_STRIDE];
  const int tid = threadIdx.x;
  const int r0  = blockIdx.x * 16;

  for (int i = tid; i < 512; i += 256) {         // 64 rows * 8 float4 (cols 0-31)
    const int row = i >> 3, off = (i & 7) << 2;
    float4 v = make_float4(0.f, 0.f, 0.f, 0.f);
    const int r = r0 + row;
    if (r < N) v = *(const float4*)(feat + (size_t)r * F_DIM + off);
    float* d = A + row * LDS_STRIDE + off;
    d[0] = v.x; d[1] = v.y; d[2] = v.z; d[3] = v.w;
  }
  __syncthreads();

  const int wave = tid >> 5, lane = tid & 31;
  const int c0 = wave * 16;
  v8f accq = wmma_gemm_tile(A, wq, c0, lane);
  v8f acck = wmma_gemm_tile(A, wk, c0, lane);

  const int col = c0 + (lane & 15);
  const int rb  = (lane >> 4) << 3;
#pragma unroll
  for (int i = 0; i < 8; ++i) {
    const int r = r0 + rb + i;
    if (r < N) {
      q[(size_t)r * F_DIM + col] = accq[i];
      k[(size_t)r * F_DIM + col] = acck[i];
    }
  }
}

// ---------------------------------------------------------------------------
// Kernel 3: 64-edge tile. The rel tile (64 x 128 fp32) is DMA'd into LDS by
// the Tensor Data Mover with hardware padding (128 DWORDs -> +4 DWORDs) so
// the LDS image has row stride 132 (bank-conflict free for WMMA A reads) and
// rows past E are zero-filled by TDM OOB handling. Then erel = rel @ w_k via
// WMMA, and e[e,h] = k[dst]·q[src] + erel·k[src] with leaky-relu + atomicMax.
// ---------------------------------------------------------------------------
__global__ void edge_kernel(const float* __restrict__ rel,
                            const int* __restrict__ src,
                            const int* __restrict__ dst,
                            const float* __restrict__ wk,
                            const float* __restrict__ q,
                            const float* __restrict__ k,
                            float* __restrict__ e_out,
                            unsigned* __restrict__ m_enc, int E) {
  extern __shared__ float smem[];
  float* A = smem;                               // TILE_M x 132
  float* R = smem + TILE_M * LDS_STRIDE;         // TILE_M x 132
  const int tid = threadIdx.x;
  const int e0  = blockIdx.x * TILE_M;

  if (tid < 32) {                                // wave 0 issues the TDM op
    const unsigned long long ga = (unsigned long long)(const void*)(rel + (size_t)e0 * F_DIM);
    const unsigned rows = (unsigned)(E - e0);    // remaining rows; TDM zero-fills OOB
    v4u g0;
    g0.x = 0x1u;                                 // count=1 valid, gather off
    g0.y = __builtin_amdgcn_groupstaticsize();   // lds_addr = dynamic LDS base (A)
    g0.z = (unsigned)ga;                         // global_addr[31:0]
    g0.w = (unsigned)(ga >> 32) | (2u << 30);    // global_addr[56:32] | type=2
    v8i g1;
    g1[0] = (int)((2u << 16)                     // data_size = 4 bytes
                | (1u << 20)                     // pad_enable
                | (6u << 22)                     // pad_interval: 128 DWORDs
                | (3u << 25));                   // pad_amount:   4 DWORDs -> stride 132
    g1[1] = (int)(128u << 16);                   // tensor_dim0 = 128
    g1[2] = (int)((rows & 0xFFFFu) << 16);       // tensor_dim1[15:0]
    g1[3] = (int)(((rows >> 16) & 0xFFFFu)       // tensor_dim1[31:16]
                | (128u << 16));                 // tile_dim0 = 128
    g1[4] = TILE_M;                              // tile_dim1 = 64 rows
    g1[5] = 128;                                 // tensor_dim0_stride = 128
    g1[6] = 0;
    g1[7] = 0;
    const v4i gz4 = {0, 0, 0, 0};                // 2D tensor: groups 2/3 unused
    const v8i gz8 = {0, 0, 0, 0, 0, 0, 0, 0};    // extra group (6-arg toolchain)
    __builtin_amdgcn_tensor_load_to_lds(g0, g1, gz4, gz4, gz8, 0);
    __builtin_amdgcn_s_wait_tensorcnt(0);
  }
  __syncthreads();

  const int wave = tid >> 5, lane = tid & 31;
  const int c0 = wave * 16;
#pragma unroll
  for (int mt = 0; mt < TILE_M / 16; ++mt) {
    v8f acc = wmma_gemm_tile(A + mt * 16 * LDS_STRIDE, wk, c0, lane);
    const int col = c0 + (lane & 15);
    const int rb  = (lane >> 4) << 3;
#pragma unroll
    for (int i = 0; i < 8; ++i)
      R[(mt * 16 + rb + i) * LDS_STRIDE + col] = acc[i];
  }
  __syncthreads();

  // 64 edges * 4 heads = 256 (edge,head) pairs -> one per thread
  const int le = tid >> 2, h = tid & 3;
  const int eg = e0 + le;
  if (eg < E) {
    const int s = src[eg], d = dst[eg];
    const float* qs = q + (size_t)s * F_DIM + h * D_DIM;
    const float* ks = k + (size_t)s * F_DIM + h * D_DIM;
    const float* kd = k + (size_t)d * F_DIM + h * D_DIM;
    const float* er = R + le * LDS_STRIDE + h * D_DIM;
    float a1 = 0.f, a2 = 0.f;
#pragma unroll 8
    for (int dd = 0; dd < D_DIM; ++dd) {
      a1 = fmaf(kd[dd], qs[dd], a1);
      a2 = fmaf(er[dd], ks[dd], a2);
    }
    float ev = a1 + a2;
    ev = ev > 0.f ? ev : 0.01f * ev;             // leaky_relu(0.01)
    e_out[(size_t)eg * H_DIM + h] = ev;
    const unsigned bits = __float_as_uint(ev);
    const unsigned enc  = (bits & 0x80000000u) ? ~bits : (bits | 0x80000000u);
    atomicMax(&m_enc[(size_t)s * H_DIM + h], enc);
  }
}

// ---------------------------------------------------------------------------
// Kernel 4: ex = exp(e - m[src]); accumulate denominators.
// ---------------------------------------------------------------------------
__global__ void softmax_num(const int* __restrict__ src,
                            const unsigned* __restrict__ m_enc,
                            float* __restrict__ e_arr,
                            float* __restrict__ denom, int total) {
  const int i = blockIdx.x * 256 + threadIdx.x;
  if (i < total) {
    const int eg = i >> 2, h = i & 3;
    const int s = src[eg];
    const unsigned u = m_enc[(size_t)s * H_DIM + h];
    const float m = (u & 0x80000000u) ? __uint_as_float(u & 0x7FFFFFFFu)
                                      : __uint_as_float(~u);
    const float ex = expf(e_arr[i] - m);
    e_arr[i] = ex;
    atomicAdd(&denom[(size_t)s * H_DIM + h], ex);
  }
}

// ---------------------------------------------------------------------------
// Kernel 5: out[e] = (1/H) * sum_h ex[e,h] / denom[src[e],h]
// ---------------------------------------------------------------------------
__global__ void finalize(const int* __restrict__ src,
                         const float* __restrict__ e_arr,
                         const float* __restrict__ denom,
                         float* __restrict__ out, int E) {
  const int eg = blockIdx.x * 256 + threadIdx.x;
  if (eg < E) {
    const int s = src[eg];
    float sum = 0.f;
#pragma unroll
    for (int h = 0; h < H_DIM; ++h)
      sum += e_arr[(size_t)eg * H_DIM + h] / denom[(size_t)s * H_DIM + h];
    out[eg] = sum * 0.25f;
  }
}

extern "C" void kernel_launch(void* const* d_in, const int* in_sizes, int n_in,
                              void* d_out, int out_size, void* d_ws, size_t ws_size,
                              hipStream_t stream) {
  const float* feat = (const float*)d_in[0];
  const float* rel  = (const float*)d_in[1];
  const int*   src  = (const int*)d_in[2];
  const int*   dst  = (const int*)d_in[3];
  const float* wq   = (const float*)d_in[4];
  const float* wk   = (const float*)d_in[5];
  float* out = (float*)d_out;

  const int N = in_sizes[0] / F_DIM;
  const int E = in_sizes[2];

  char* ws = (char*)d_ws;
  float*    q      = (float*)ws;     ws += (size_t)N * F_DIM * sizeof(float);
  float*    k      = (float*)ws;     ws += (size_t)N * F_DIM * sizeof(float);
  float*    e_arr  = (float*)ws;     ws += (size_t)E * H_DIM * sizeof(float);
  unsigned* m_enc  = (unsigned*)ws;  ws += (size_t)N * H_DIM * sizeof(unsigned);
  float*    denom  = (float*)ws;

  const size_t edge_lds = (size_t)2 * TILE_M * LDS_STRIDE * sizeof(float); // 67.6 KB

  init_seg<<<(N * H_DIM + 255) / 256, 256, 0, stream>>>(m_enc, denom, N * H_DIM);
  qk_gemm<<<(N + 15) / 16, 256, 0, stream>>>(feat, wq, wk, q, k, N);
  edge_kernel<<<(E + TILE_M - 1) / TILE_M, 256, edge_lds, stream>>>(
      rel, src, dst, wk, q, k, e_arr, m_enc, E);
  softmax_num<<<(E * H_DIM + 255) / 256, 256, 0, stream>>>(src, m_enc, e_arr, denom, E * H_DIM);
  finalize<<<(E + 255) / 256, 256, 0, stream>>>(src, e_arr, denom, out, E);
}